// EncoderBUAttention_22874995818571
// MI455X (gfx1250) — compile-verified
//
#include <hip/hip_runtime.h>
#include <hip/hip_bf16.h>

// ---------------------------------------------------------------------------
// Faster-RCNN box-head postprocess for MI455X (gfx1250, wave32).
// Stage 1: softmax (denominator via V_WMMA_F32_16X16X4_F32 row-sum) + decode
// Stage 2: per-image top-2048 via score histogram + LDS bitonic sort
// Stage 3: 2048x2048 IoU suppression bitmask (parallel)
// Stage 4: single-wave greedy NMS scan over the bitmask (+ global_prefetch)
// Stage 5: feature gather via Tensor Data Mover (tensor_load_to_lds)
// ---------------------------------------------------------------------------

#define BATCH   8
#define NPROP   1000
#define NCLS    91
#define NDET    90          // NCLS - 1 (background dropped)
#define NCAND   (NPROP * NDET)   // 90000
#define DFEAT   1024
#define KTOP    2048
#define DETS    36
#define NMS_T   0.5f
#define SCORE_T 0.2f
#define MIN_SZ  0.01f
#define XFORM_CLIP 4.135166556742356f   // log(1000/16)

typedef __attribute__((ext_vector_type(2))) float        v2f;
typedef __attribute__((ext_vector_type(8))) float        v8f;
typedef __attribute__((ext_vector_type(4))) unsigned int v4u;
typedef __attribute__((ext_vector_type(8))) int          v8i;
typedef __attribute__((ext_vector_type(4))) int          v4i;

#if defined(__has_builtin)
#if __has_builtin(__builtin_amdgcn_tensor_load_to_lds) && \
    __has_builtin(__builtin_amdgcn_s_wait_tensorcnt)
#define HAVE_TDM 1
#endif
#endif

__device__ __forceinline__ float sel8(v8f a, int i) {
    // dynamic extract from 8-wide accumulator via select tree
    return (i & 4) ? ((i & 2) ? ((i & 1) ? a[7] : a[6]) : ((i & 1) ? a[5] : a[4]))
                   : ((i & 2) ? ((i & 1) ? a[3] : a[2]) : ((i & 1) ? a[1] : a[0]));
}

// ---------------------------------------------------------------------------
// Kernel 1: per-proposal softmax + box decode + clip + validity mask.
// One wave handles 16 proposal rows; softmax denominator computed with
// V_WMMA_F32_16X16X4_F32 (A = exp(logits - rowmax) 16x4 tile, B = ones).
// Main K-loop is branch-free (EXEC stays all-ones); tail chunk peeled.
// ---------------------------------------------------------------------------
__global__ __launch_bounds__(256) void k_decode(
    const float* __restrict__ logits,   // [B, N, C]
    const float* __restrict__ reg,      // [B, N, C*4]
    const float* __restrict__ props,    // [B, N, 4]
    const int*   __restrict__ imghw,    // [B, 2] (h, w)
    float*       __restrict__ scoresOut,// [B, N*90]  (-1 if invalid)
    float4*      __restrict__ boxesOut) // [B, N*90]
{
    const int b    = blockIdx.x;
    const int lane = threadIdx.x & 31;
    const int wv   = threadIdx.x >> 5;
    const int half = lane >> 4;         // 0: cols k,k+1   1: cols k+2,k+3
    const int r    = lane & 15;         // matrix row (proposal within wave)
    const int row  = blockIdx.y * 128 + wv * 16 + r;
    const int rowc = (row < NPROP) ? row : (NPROP - 1);

    const float* lrow = logits + ((size_t)b * NPROP + rowc) * NCLS;

    // ---- phase A: row max (2 lanes per row split the 91 classes) ----
    float m = -INFINITY;
    {
        const int c0 = half ? 46 : 0;
        const int c1 = half ? 91 : 46;
        for (int c = c0; c < c1; ++c) m = fmaxf(m, lrow[c]);
        m = fmaxf(m, __shfl_xor(m, 16, 32));
    }

    // ---- phase B: sum(exp) via f32 WMMA, K-loop over 91 classes ----
    v8f acc = {};                       // C/D accumulator (16x16 f32)
    v2f bones; bones[0] = 1.0f; bones[1] = 1.0f;   // B = all-ones 4x16
    const float* lp = lrow + 2 * half;
    #pragma unroll 1
    for (int k0 = 0; k0 < 88; k0 += 4) {           // classes 0..87: no bounds checks
        v2f a;
        a[0] = __expf(lp[k0]     - m);
        a[1] = __expf(lp[k0 + 1] - m);
        acc = __builtin_amdgcn_wmma_f32_16x16x4_f32(
                  false, a, false, bones, (short)0, acc, false, false);
    }
    {   // peeled tail k0 = 88: classes 88..90 (class 91 does not exist -> 0)
        const int ca = 88 + 2 * half;
        const int cb = ca + 1;
        const float lvb = lrow[(cb < NCLS) ? cb : (NCLS - 1)];
        v2f a;
        a[0] = __expf(lrow[ca] - m);
        a[1] = (cb < NCLS) ? __expf(lvb - m) : 0.0f;
        acc = __builtin_amdgcn_wmma_f32_16x16x4_f32(
                  false, a, false, bones, (short)0, acc, false, false);
    }
    // every column of D equals the row sum; lanes 0-15 hold rows 0..7 in
    // acc[0..7], lanes 16-31 hold rows 8..15.
    const float own = sel8(acc, r & 7);
    const float oth = __shfl_xor(own, 16, 32);
    const float denom = ((r >> 3) == half) ? own : oth;
    const float inv = 1.0f / denom;

    // ---- phase C: probabilities + box decode + clip + validity ----
    if (row < NPROP) {
        const float* rrow = reg   + ((size_t)b * NPROP + row) * (NCLS * 4);
        const float* p    = props + ((size_t)b * NPROP + row) * 4;
        const float pw_ = p[2] - p[0];
        const float ph_ = p[3] - p[1];
        const float pcx = p[0] + 0.5f * pw_;
        const float pcy = p[1] + 0.5f * ph_;
        const float H = (float)imghw[b * 2 + 0];
        const float W = (float)imghw[b * 2 + 1];

        const int cs = half ? 46 : 1;   // skip background class 0
        const int ce = half ? 91 : 46;
        for (int c = cs; c < ce; ++c) {
            const float prob = __expf(lrow[c] - m) * inv;
            const float dx = rrow[c * 4 + 0] * 0.1f;       // / WX
            const float dy = rrow[c * 4 + 1] * 0.1f;       // / WY
            const float dw = fminf(rrow[c * 4 + 2] * 0.2f, XFORM_CLIP);
            const float dh = fminf(rrow[c * 4 + 3] * 0.2f, XFORM_CLIP);
            const float cx = dx * pw_ + pcx;
            const float cy = dy * ph_ + pcy;
            const float bw = __expf(dw) * pw_;
            const float bh = __expf(dh) * ph_;
            const float x1 = fminf(fmaxf(cx - 0.5f * bw, 0.0f), W);
            const float y1 = fminf(fmaxf(cy - 0.5f * bh, 0.0f), H);
            const float x2 = fminf(fmaxf(cx + 0.5f * bw, 0.0f), W);
            const float y2 = fminf(fmaxf(cy + 0.5f * bh, 0.0f), H);
            const bool valid = (prob > SCORE_T) &&
                               ((x2 - x1) >= MIN_SZ) && ((y2 - y1) >= MIN_SZ);
            const size_t si = ((size_t)b * NPROP + row) * NDET + (c - 1);
            scoresOut[si] = valid ? prob : -1.0f;
            boxesOut[si]  = make_float4(x1, y1, x2, y2);
        }
    }
}

// ---------------------------------------------------------------------------
// Kernel 2: per-image top-2048 selection (score histogram cut + compaction)
// followed by LDS bitonic sort (descending score, ascending index on ties).
// ---------------------------------------------------------------------------
__global__ __launch_bounds__(1024) void k_topk(
    const float*  __restrict__ scores,  // [B, 90000]
    const float4* __restrict__ boxes,   // [B, 90000]
    float*        __restrict__ topS,    // [B, 2048]
    int*          __restrict__ topI,    // [B, 2048]
    float4*       __restrict__ topB)    // [B, 2048]
{
    __shared__ unsigned hist[1024];
    __shared__ int      suf[1024];
    __shared__ float    ss[KTOP];
    __shared__ int      si[KTOP];
    __shared__ int      sCut, sCntHi, sCntMid, sFilled;

    const int b   = blockIdx.x;
    const int tid = threadIdx.x;
    const int nt  = blockDim.x;
    const float* sc = scores + (size_t)b * NCAND;

    for (int i = tid; i < 1024; i += nt) hist[i] = 0;
    if (tid == 0) { sCut = -1; sCntHi = 0; sCntMid = 0; }
    __syncthreads();

    // pass 1: histogram over (0.2, 1.0]
    for (int i = tid; i < NCAND; i += nt) {
        const float s = sc[i];
        if (s > SCORE_T) {
            int bin = (int)((s - SCORE_T) * (1024.0f / (1.0f - SCORE_T)));
            bin = min(max(bin, 0), 1023);
            atomicAdd(&hist[bin], 1u);
        }
    }
    __syncthreads();
    suf[tid] = (int)hist[tid];
    __syncthreads();
    // suffix scan (Hillis-Steele)
    for (int off = 1; off < 1024; off <<= 1) {
        int v = (tid + off < 1024) ? suf[tid + off] : 0;
        __syncthreads();
        suf[tid] += v;
        __syncthreads();
    }
    // cut bin: highest b with suf[b] >= KTOP (unique since suf is monotone)
    {
        const int above = (tid < 1023) ? suf[tid + 1] : 0;
        if (suf[tid] >= KTOP && above < KTOP) sCut = tid;
    }
    __syncthreads();
    const int cutb = sCut;
    const int nA = (cutb < 0) ? 0 : ((cutb < 1023) ? suf[cutb + 1] : 0);

    // pass 2: compact candidates into LDS
    for (int i = tid; i < NCAND; i += nt) {
        const float s = sc[i];
        if (s > SCORE_T) {
            int bin = (int)((s - SCORE_T) * (1024.0f / (1.0f - SCORE_T)));
            bin = min(max(bin, 0), 1023);
            int slot = -1;
            if (cutb < 0 || bin > cutb)  slot = atomicAdd(&sCntHi, 1);
            else if (bin == cutb)        slot = nA + atomicAdd(&sCntMid, 1);
            if (slot >= 0 && slot < KTOP) { ss[slot] = s; si[slot] = i; }
        }
    }
    __syncthreads();
    if (tid == 0) {
        int midTaken = sCntMid; if (midTaken > KTOP - nA) midTaken = KTOP - nA;
        int f = sCntHi + ((cutb < 0) ? 0 : midTaken);
        sFilled = (f > KTOP) ? KTOP : f;
    }
    __syncthreads();
    for (int i = tid; i < KTOP; i += nt)
        if (i >= sFilled) { ss[i] = -INFINITY; si[i] = 0x7fffffff; }
    __syncthreads();

    // bitonic sort: descending score, ascending index on ties (deterministic)
    for (int k = 2; k <= KTOP; k <<= 1) {
        for (int j = k >> 1; j > 0; j >>= 1) {
            for (int t = tid; t < KTOP; t += nt) {
                const int p = t ^ j;
                if (p > t) {
                    const float a = ss[t], c = ss[p];
                    const int   ia = si[t], ic = si[p];
                    const bool before = (a > c) || (a == c && ia < ic);
                    const bool dirDesc = ((t & k) == 0);
                    const bool sw = dirDesc ? !before : before;
                    if (sw) { ss[t] = c; ss[p] = a; si[t] = ic; si[p] = ia; }
                }
            }
            __syncthreads();
        }
    }

    // emit sorted candidates + gathered boxes
    for (int t = tid; t < KTOP; t += nt) {
        const float s = ss[t];
        const int   id = si[t];
        topS[b * KTOP + t] = s;
        topI[b * KTOP + t] = (s > -1e30f) ? id : 0;
        topB[b * KTOP + t] = (s > -1e30f) ? boxes[(size_t)b * NCAND + id]
                                          : make_float4(0.f, 0.f, 0.f, 0.f);
    }
}

// ---------------------------------------------------------------------------
// Kernel 3: suppression bitmask.  bit t of mask[b][i][w] = (IoU(i, w*32+t)
// > 0.5 and j > i).  One thread per (b, i, w).
// ---------------------------------------------------------------------------
__global__ __launch_bounds__(256) void k_mask(
    const float4* __restrict__ topB,    // [B, 2048]
    unsigned*     __restrict__ mask)    // [B, 2048, 64]
{
    const int idx = blockIdx.x * blockDim.x + threadIdx.x;   // B*2048*64
    const int b   = idx >> 17;
    const int rem = idx & 131071;
    const int i   = rem >> 6;
    const int w   = rem & 63;
    const float4* tb = topB + (size_t)b * KTOP;

    const float4 bi = tb[i];
    const float areai = (bi.z - bi.x) * (bi.w - bi.y);
    unsigned bits = 0;
    const int j0 = w << 5;
    #pragma unroll 4
    for (int t = 0; t < 32; ++t) {
        const int j = j0 + t;
        if (j > i) {
            const float4 bj = tb[j];
            const float xx1 = fmaxf(bi.x, bj.x);
            const float yy1 = fmaxf(bi.y, bj.y);
            const float xx2 = fminf(bi.z, bj.z);
            const float yy2 = fminf(bi.w, bj.w);
            const float iw = fmaxf(xx2 - xx1, 0.0f);
            const float ih = fmaxf(yy2 - yy1, 0.0f);
            const float inter = iw * ih;
            const float areaj = (bj.z - bj.x) * (bj.w - bj.y);
            const float uni = areai + areaj - inter;
            const float iou = (uni > 0.0f) ? inter / uni : 0.0f;
            if (iou > NMS_T) bits |= (1u << t);
        }
    }
    mask[idx] = bits;
}

// ---------------------------------------------------------------------------
// Kernel 4: sequential greedy scan, one wave per image.  "removed" bitmask
// (2048 bits) lives in 2 VGPRs per lane; rows of the suppression mask are
// ORed in only for kept candidates.  Rows are software-prefetched 8 ahead
// (global_prefetch_b8) to hide L2 latency behind the register-only skips.
// ---------------------------------------------------------------------------
__global__ __launch_bounds__(32) void k_scan(
    const float*    __restrict__ topS,
    const int*      __restrict__ topI,
    const unsigned* __restrict__ mask,
    int*            __restrict__ selProp)   // [B, 36], -1 = empty slot
{
    const int b    = blockIdx.x;
    const int lane = threadIdx.x;
    const float* s = topS + (size_t)b * KTOP;

    // init removed = ~valid (invalid candidates can never be kept)
    unsigned r0 = 0, r1 = 0;
    for (int g = 0; g < 64; ++g) {
        const float v = s[(g << 5) + lane];
        const unsigned bad = (unsigned)__ballot(!(v > -1e30f));
        if ((g >> 1) == lane) { if (g & 1) r1 = bad; else r0 = bad; }
    }

    const uint2* mrow = (const uint2*)(mask + (size_t)b * KTOP * 64);
    int count = 0;
    for (int i = 0; i < KTOP && count < DETS; ++i) {
        // speculative prefetch of the mask row 8 iterations ahead
        if (i + 8 < KTOP)
            __builtin_prefetch((const void*)(mrow + (size_t)(i + 8) * 32 + lane), 0, 1);

        const int word  = i >> 5;
        const int owner = word >> 1;
        const unsigned selw = (word & 1) ? r1 : r0;
        const unsigned bits = __shfl(selw, owner, 32);
        if (!((bits >> (i & 31)) & 1u)) {
            if (lane == 0)
                selProp[b * DETS + count] = topI[b * KTOP + i] / NDET;
            ++count;
            const uint2 mr = mrow[(size_t)i * 32 + lane];
            r0 |= mr.x;
            r1 |= mr.y;
        }
    }
    if (lane == 0)
        for (int k = count; k < DETS; ++k) selProp[b * DETS + k] = -1;
}

// ---------------------------------------------------------------------------
// Kernel 5: feature gather.  One wave per output row.  The 4 KB feature row
// is DMA'd global->LDS by the Tensor Data Mover (tensor_load_to_lds, D#
// describing a 1x1024 f32 tile), synchronized with s_wait_tensorcnt, then
// streamed LDS->global.  Empty slots (-1) are zero-filled.
// ---------------------------------------------------------------------------
__global__ __launch_bounds__(32) void k_gather(
    const float* __restrict__ feats,    // [B, N, 1024]
    const int*   __restrict__ selProp,  // [B, 36]
    float4*      __restrict__ out)      // [B, 36, 256] as float4
{
    __shared__ float buf[DFEAT];        // 4 KB staging tile
    const int t    = blockIdx.x;        // b*36 + s
    const int b    = t / DETS;
    const int lane = threadIdx.x;
    const int sel  = selProp[t];
    float4* orow = out + (size_t)t * (DFEAT / 4);

    if (sel >= 0) {
#if defined(HAVE_TDM)
        const unsigned long long gaddr = (unsigned long long)(uintptr_t)
            (feats + ((size_t)b * NPROP + sel) * DFEAT);
        const unsigned ldsOff = (unsigned)(uintptr_t)&buf[0];
        // ---- D# group 0: count=1, lds_addr, global_addr, type=2 ----
        v4u g0;
        g0[0] = 1u;
        g0[1] = ldsOff;
        g0[2] = (unsigned)(gaddr & 0xffffffffu);
        g0[3] = (unsigned)((gaddr >> 32) & 0x01ffffffu) | (2u << 30);
        // ---- D# group 1: data_size=4B, tensor 1024x1, tile 1024x1 ----
        v8i g1;
        g1[0] = (int)(2u << 16);                  // data_size = 4 bytes
        g1[1] = (int)((unsigned)DFEAT << 16);     // tensor_dim0[15:0] @ bits 63:48
        g1[2] = (int)(1u << 16);                  // tensor_dim1 = 1
        g1[3] = (int)((unsigned)DFEAT << 16);     // tile_dim0 @ bits 127:112
        g1[4] = 1;                                // tile_dim1 = 1
        g1[5] = DFEAT;                            // tensor_dim0_stride
        g1[6] = 0;
        g1[7] = 0;
        v4i gz;  gz[0] = 0; gz[1] = 0; gz[2] = 0; gz[3] = 0;      // groups 2/3: unused
        v8i gz8; gz8[0] = 0; gz8[1] = 0; gz8[2] = 0; gz8[3] = 0;  // 6-arg form extra group
        gz8[4] = 0; gz8[5] = 0; gz8[6] = 0; gz8[7] = 0;
        __builtin_amdgcn_tensor_load_to_lds(g0, g1, gz, gz, gz8, 0);
        __builtin_amdgcn_s_wait_tensorcnt((short)0);
        // volatile: LDS was written by the DMA engine, not by this wave
        const volatile float* vf = (const volatile float*)buf;
        for (int q = lane * 4; q < DFEAT; q += 128)
            orow[q >> 2] = make_float4(vf[q], vf[q + 1], vf[q + 2], vf[q + 3]);
#else
        const float4* frow = (const float4*)(feats + ((size_t)b * NPROP + sel) * DFEAT);
        for (int q = lane; q < DFEAT / 4; q += 32) orow[q] = frow[q];
#endif
    } else {
        const float4 z = make_float4(0.f, 0.f, 0.f, 0.f);
        for (int q = lane; q < DFEAT / 4; q += 32) orow[q] = z;
    }
}

// ---------------------------------------------------------------------------
extern "C" void kernel_launch(void* const* d_in, const int* in_sizes, int n_in,
                              void* d_out, int out_size, void* d_ws, size_t ws_size,
                              hipStream_t stream) {
    (void)in_sizes; (void)n_in; (void)out_size; (void)ws_size;
    const float* logits = (const float*)d_in[0];   // [8,1000,91]
    const float* reg    = (const float*)d_in[1];   // [8,1000,364]
    const float* props  = (const float*)d_in[2];   // [8,1000,4]
    const float* feats  = (const float*)d_in[3];   // [8,1000,1024]
    const int*   imghw  = (const int*)  d_in[4];   // [8,2]

    // workspace layout (bytes), all 16B aligned; total ~19 MB
    char* ws = (char*)d_ws;
    float*    scores  = (float*)   (ws + 0);           //  2,880,000 B
    float4*   boxes   = (float4*)  (ws + 2880000);     // 11,520,000 B
    float*    topS    = (float*)   (ws + 14400000);    //     65,536 B
    int*      topI    = (int*)     (ws + 14465536);    //     65,536 B
    float4*   topB    = (float4*)  (ws + 14531072);    //    262,144 B
    unsigned* mask    = (unsigned*)(ws + 14793216);    //  4,194,304 B
    int*      selProp = (int*)     (ws + 18987520);    //      1,152 B

    dim3 g1(BATCH, 8);
    k_decode<<<g1, 256, 0, stream>>>(logits, reg, props, imghw, scores, boxes);
    k_topk  <<<BATCH, 1024, 0, stream>>>(scores, boxes, topS, topI, topB);
    k_mask  <<<(BATCH * KTOP * 64) / 256, 256, 0, stream>>>(topB, mask);
    k_scan  <<<BATCH, 32, 0, stream>>>(topS, topI, mask, selProp);
    k_gather<<<BATCH * DETS, 32, 0, stream>>>(feats, selProp, (float4*)d_out);
}